// Adjoint_46273977647342
// MI455X (gfx1250) — compile-verified
//
#include <hip/hip_runtime.h>
#include <hip/hip_bf16.h>
#include <math.h>

typedef __attribute__((ext_vector_type(2))) float v2f;
typedef __attribute__((ext_vector_type(8))) float v8f;

#define M_PTS 8192
#define NXY   192
#define NCOIL 8

#define WMMA_F32(A, B, C) \
    __builtin_amdgcn_wmma_f32_16x16x4_f32(false, (A), false, (B), (short)0, (C), false, false)

// ---------------------------------------------------------------------------
// Kernel 1: build phase tables and density-compensated k-space.
//   ExT[x][m]       planar re/im  (x-major: A rows have contiguous K=m)
//   Eyc[m][y]       interleaved (re,im) float2  (m-major: B rows contiguous)
//   Wc [c][m]       interleaved (re,im) float2
// ---------------------------------------------------------------------------
__global__ __launch_bounds__(256)
void nufft_tables_kernel(const float* __restrict__ kr, const float* __restrict__ ki,
                         const float* __restrict__ traj, const float* __restrict__ dc,
                         float* __restrict__ exre, float* __restrict__ exim,
                         v2f* __restrict__ eyc, v2f* __restrict__ wc)
{
    const int id  = blockIdx.x * blockDim.x + threadIdx.x;
    const int NEX = NXY * M_PTS;
    if (id < NEX) {
        const int x = id / M_PTS;
        const int m = id - x * M_PTS;
        const float ph = traj[m] * (float)(x - NXY / 2);         // kx = traj[0*M+m]
        float s, c;
        __sincosf(ph, &s, &c);
        exre[id] = c;
        exim[id] = s;
    } else if (id < 2 * NEX) {
        const int j = id - NEX;                                  // j = m*NXY + y
        const int m = j / NXY;
        const int y = j - m * NXY;
        const float ph = traj[M_PTS + m] * (float)(y - NXY / 2); // ky = traj[1*M+m]
        float s, c;
        __sincosf(ph, &s, &c);
        v2f e; e.x = c; e.y = s;
        eyc[j] = e;
    } else if (id < 2 * NEX + NCOIL * M_PTS) {
        const int j = id - 2 * NEX;                              // j = c*M + m
        const float d = dc[j & (M_PTS - 1)];
        v2f w; w.x = kr[j] * d; w.y = ki[j] * d;
        wc[j] = w;
    }
}

// ---------------------------------------------------------------------------
// Kernel 2: per-coil complex GEMM  recon_c = ExT(192xK) * B_c(Kx192), K=8192,
// on the fp32 WMMA pipe. Each wave owns a 32x32 output tile = 2x2 WMMA tiles
// x 4 accumulators (rr, ii, ri, ir) -> 16 v_wmma per K=4 step against only
// ~10 load instructions. B_c[m,y] = W[c,m]*Ey[m,y] built in registers.
// Grid: 288 wave-tiles (8 coils * 6 * 6) = 36 blocks of 8 waves.
// ---------------------------------------------------------------------------
__global__ __launch_bounds__(256)
void nufft_wmma_gemm_kernel(const float* __restrict__ exre, const float* __restrict__ exim,
                            const v2f* __restrict__ eyc,  const v2f* __restrict__ wc,
                            float* __restrict__ rre, float* __restrict__ rim)
{
    const int wave = threadIdx.x >> 5;
    const int lane = threadIdx.x & 31;
    const int g    = blockIdx.x * 8 + wave;          // 0..287
    const int c    = g / 36;
    const int tile = g - c * 36;                     // 0..35
    const int tx   = tile % 6;
    const int ty   = tile / 6;
    const int x0   = tx * 32;
    const int y0   = ty * 32;

    const int lm    = lane & 15;          // row (A) / col (B) within 16-tile
    const int khalf = (lane >> 4) * 2;    // lanes 0-15: K+{0,1}; 16-31: K+{2,3}

    const float* __restrict__ ar0 = exre + (size_t)(x0 + lm)      * M_PTS + khalf;
    const float* __restrict__ ai0 = exim + (size_t)(x0 + lm)      * M_PTS + khalf;
    const float* __restrict__ ar1 = exre + (size_t)(x0 + 16 + lm) * M_PTS + khalf;
    const float* __restrict__ ai1 = exim + (size_t)(x0 + 16 + lm) * M_PTS + khalf;
    const v2f*   __restrict__ wrow = wc + (size_t)c * M_PTS;

    v8f rr00 = {}, ii00 = {}, ri00 = {}, ir00 = {};
    v8f rr01 = {}, ii01 = {}, ri01 = {}, ir01 = {};
    v8f rr10 = {}, ii10 = {}, ri10 = {}, ir10 = {};
    v8f rr11 = {}, ii11 = {}, ri11 = {}, ir11 = {};

    for (int k = 0; k < M_PTS; k += 4) {
        // A fragments (16x4 f32), two row tiles, planar re/im.
        v2f a_re0 = *(const v2f*)(ar0 + k);
        v2f a_im0 = *(const v2f*)(ai0 + k);
        v2f a_re1 = *(const v2f*)(ar1 + k);
        v2f a_im1 = *(const v2f*)(ai1 + k);

        // B fragments: rows m0=k+khalf, m1=m0+1; two column tiles.
        const int m0 = k + khalf;
        const int m1 = m0 + 1;
        const v2f w0  = wrow[m0];                       // {wre, wim}
        const v2f w1  = wrow[m1];
        const v2f e00 = eyc[(size_t)m0 * NXY + y0 + lm];        // col tile 0
        const v2f e01 = eyc[(size_t)m0 * NXY + y0 + 16 + lm];   // col tile 1
        const v2f e10 = eyc[(size_t)m1 * NXY + y0 + lm];
        const v2f e11 = eyc[(size_t)m1 * NXY + y0 + 16 + lm];

        v2f b_re0, b_im0, b_re1, b_im1;
        b_re0.x = w0.x * e00.x - w0.y * e00.y;
        b_im0.x = w0.x * e00.y + w0.y * e00.x;
        b_re0.y = w1.x * e10.x - w1.y * e10.y;
        b_im0.y = w1.x * e10.y + w1.y * e10.x;
        b_re1.x = w0.x * e01.x - w0.y * e01.y;
        b_im1.x = w0.x * e01.y + w0.y * e01.x;
        b_re1.y = w1.x * e11.x - w1.y * e11.y;
        b_im1.y = w1.x * e11.y + w1.y * e11.x;

        rr00 = WMMA_F32(a_re0, b_re0, rr00);
        ii00 = WMMA_F32(a_im0, b_im0, ii00);
        ri00 = WMMA_F32(a_re0, b_im0, ri00);
        ir00 = WMMA_F32(a_im0, b_re0, ir00);

        rr01 = WMMA_F32(a_re0, b_re1, rr01);
        ii01 = WMMA_F32(a_im0, b_im1, ii01);
        ri01 = WMMA_F32(a_re0, b_im1, ri01);
        ir01 = WMMA_F32(a_im0, b_re1, ir01);

        rr10 = WMMA_F32(a_re1, b_re0, rr10);
        ii10 = WMMA_F32(a_im1, b_im0, ii10);
        ri10 = WMMA_F32(a_re1, b_im0, ri10);
        ir10 = WMMA_F32(a_im1, b_re0, ir10);

        rr11 = WMMA_F32(a_re1, b_re1, rr11);
        ii11 = WMMA_F32(a_im1, b_im1, ii11);
        ri11 = WMMA_F32(a_re1, b_im1, ri11);
        ir11 = WMMA_F32(a_im1, b_re1, ir11);
    }

    // D layout (16x16 f32): VGPR v, lanes 0-15 -> row v; lanes 16-31 -> row v+8.
    const int rbase = (lane >> 4) * 8;
    float* __restrict__ outre = rre + (size_t)c * NXY * NXY;
    float* __restrict__ outim = rim + (size_t)c * NXY * NXY;
#pragma unroll
    for (int v = 0; v < 8; ++v) {
        const size_t rlo = (size_t)(x0 + rbase + v) * NXY;
        const size_t rhi = (size_t)(x0 + 16 + rbase + v) * NXY;
        const int c0 = y0 + lm;
        const int c1 = y0 + 16 + lm;
        outre[rlo + c0] = rr00[v] - ii00[v];
        outim[rlo + c0] = ri00[v] + ir00[v];
        outre[rlo + c1] = rr01[v] - ii01[v];
        outim[rlo + c1] = ri01[v] + ir01[v];
        outre[rhi + c0] = rr10[v] - ii10[v];
        outim[rhi + c0] = ri10[v] + ir10[v];
        outre[rhi + c1] = rr11[v] - ii11[v];
        outim[rhi + c1] = ri11[v] + ir11[v];
    }
}

// ---------------------------------------------------------------------------
// Kernel 3: coil combine with conj(smaps) + magnitude.
// ---------------------------------------------------------------------------
__global__ __launch_bounds__(256)
void nufft_combine_kernel(const float* __restrict__ rre, const float* __restrict__ rim,
                          const float* __restrict__ sre, const float* __restrict__ sim,
                          float* __restrict__ out)
{
    const int id = blockIdx.x * blockDim.x + threadIdx.x;
    if (id >= NXY * NXY) return;
    float ar = 0.f, ai = 0.f;
#pragma unroll
    for (int c = 0; c < NCOIL; ++c) {
        const size_t o = (size_t)c * NXY * NXY + id;
        const float rr = rre[o], ri = rim[o];
        const float sr = sre[o], si = sim[o];
        ar += rr * sr + ri * si;            // Re(r * conj(s))
        ai += ri * sr - rr * si;            // Im(r * conj(s))
    }
    out[id] = sqrtf(ar * ar + ai * ai);
}

// ---------------------------------------------------------------------------
extern "C" void kernel_launch(void* const* d_in, const int* in_sizes, int n_in,
                              void* d_out, int out_size, void* d_ws, size_t ws_size,
                              hipStream_t stream)
{
    const float* kr   = (const float*)d_in[0];  // kspace_real  [1,8,8192]
    const float* ki   = (const float*)d_in[1];  // kspace_imag  [1,8,8192]
    const float* traj = (const float*)d_in[2];  // trajectory   [2,8192]
    const float* sre  = (const float*)d_in[3];  // smaps_real   [1,8,192,192]
    const float* sim  = (const float*)d_in[4];  // smaps_imag   [1,8,192,192]
    const float* dc   = (const float*)d_in[5];  // density_comp [8192]

    float* ws = (float*)d_ws;
    const size_t NEX = (size_t)NXY * M_PTS;     // 1,572,864
    float* exre = ws;                  ws += NEX;
    float* exim = ws;                  ws += NEX;
    v2f*   eyc  = (v2f*)ws;            ws += 2 * NEX;
    v2f*   wcp  = (v2f*)ws;            ws += 2 * (size_t)NCOIL * M_PTS;
    float* rre  = ws;                  ws += (size_t)NCOIL * NXY * NXY;
    float* rim  = ws;                  // total ~28 MB (fits the 192 MB L2)

    // 1) phase tables + density-compensated k-space
    {
        const int total  = 2 * (int)NEX + NCOIL * M_PTS;   // 3,211,264
        const int blocks = (total + 255) / 256;
        nufft_tables_kernel<<<blocks, 256, 0, stream>>>(kr, ki, traj, dc,
                                                        exre, exim, eyc, wcp);
    }
    // 2) per-coil complex GEMM on fp32 WMMA (288 wave-tiles, 36 blocks)
    nufft_wmma_gemm_kernel<<<36, 256, 0, stream>>>(exre, exim, eyc, wcp, rre, rim);
    // 3) coil combine + magnitude
    nufft_combine_kernel<<<(NXY * NXY + 255) / 256, 256, 0, stream>>>(rre, rim,
                                                                      sre, sim,
                                                                      (float*)d_out);
}